// Forest_83193516523594
// MI455X (gfx1250) — compile-verified
//
#include <hip/hip_runtime.h>

#define BATCH 8192
#define NTREES 4
#define NNODES 2048
#define SLOPE 0.01f
#define EPSV 1e-5f

typedef __attribute__((ext_vector_type(2))) float v2f;
typedef __attribute__((ext_vector_type(8))) float v8f;

__device__ __forceinline__ float leakyf(float x) {
    return x >= 0.0f ? x : SLOPE * x;
}

__device__ __forceinline__ float sigmoidf(float x) {
    return __builtin_amdgcn_rcpf(1.0f + __expf(-x));
}

// ---------------- zero workspace accumulators ----------------
__global__ void k_zero(float* ws, int n) {
    int i = blockIdx.x * blockDim.x + threadIdx.x;
    if (i < n) ws[i] = 0.0f;
}

// ---------------- BN1 batch statistics ----------------
__global__ __launch_bounds__(256) void k_bn1_stats(const float* __restrict__ xb,
                                                   const float* __restrict__ w1,
                                                   const float* __restrict__ b1,
                                                   float* __restrict__ stats1) {
    __shared__ float ls[32];
    if (threadIdx.x < 32) ls[threadIdx.x] = 0.0f;
    __syncthreads();
    int b = blockIdx.x * 256 + threadIdx.x;
    float x = xb[b];
    for (int c = 0; c < 16; ++c) {
        float v = leakyf(fmaf(x, w1[c], b1[c]));
        atomicAdd(&ls[c], v);
        atomicAdd(&ls[16 + c], v * v);
    }
    __syncthreads();
    if (threadIdx.x < 32) atomicAdd(&stats1[threadIdx.x], ls[threadIdx.x]);
}

// ---------------- fold BN1 + pre_w2 into affine ----------------
__global__ void k_consts1(const float* __restrict__ stats1,
                          const float* __restrict__ g1,
                          const float* __restrict__ be1,
                          const float* __restrict__ w2,
                          const float* __restrict__ b2,
                          float* __restrict__ Avec,
                          float* __restrict__ C0) {
    if (threadIdx.x == 0 && blockIdx.x == 0) {
        float c0 = b2[0];
        for (int c = 0; c < 16; ++c) {
            float m   = stats1[c] * (1.0f / BATCH);
            float var = stats1[16 + c] * (1.0f / BATCH) - m * m;
            float is  = rsqrtf(var + EPSV);
            float Ac  = w2[c] * g1[c] * is;
            Avec[c]   = Ac;
            c0 += w2[c] * be1[c] - Ac * m;
        }
        C0[0] = c0;
    }
}

// ---------------- layer 2 pre-activation + BN2 stats ----------------
__global__ __launch_bounds__(256) void k_bn2_stats(const float* __restrict__ xb,
                                                   const float* __restrict__ w1,
                                                   const float* __restrict__ b1,
                                                   const float* __restrict__ Avec,
                                                   const float* __restrict__ C0,
                                                   float* __restrict__ t2,
                                                   float* __restrict__ stats2) {
    __shared__ float ls[2];
    if (threadIdx.x < 2) ls[threadIdx.x] = 0.0f;
    __syncthreads();
    int b = blockIdx.x * 256 + threadIdx.x;
    float x = xb[b];
    float z = C0[0];
    for (int c = 0; c < 16; ++c) {
        float v = leakyf(fmaf(x, w1[c], b1[c]));
        z = fmaf(Avec[c], v, z);
    }
    z = leakyf(z);
    t2[b] = z;
    atomicAdd(&ls[0], z);
    atomicAdd(&ls[1], z * z);
    __syncthreads();
    if (threadIdx.x < 2) atomicAdd(&stats2[threadIdx.x], ls[threadIdx.x]);
}

__global__ void k_consts2(const float* __restrict__ stats2,
                          const float* __restrict__ g2,
                          const float* __restrict__ be2,
                          float* __restrict__ c2) {
    if (threadIdx.x == 0 && blockIdx.x == 0) {
        float m   = stats2[0] * (1.0f / BATCH);
        float var = stats2[1] * (1.0f / BATCH) - m * m;
        float is  = rsqrtf(var + EPSV);
        float s   = is * g2[0];
        c2[0] = s;
        c2[1] = be2[0] - m * s;
    }
}

__global__ __launch_bounds__(256) void k_xs(const float* __restrict__ t2,
                                            const float* __restrict__ c2,
                                            float* __restrict__ xs) {
    int b = blockIdx.x * 256 + threadIdx.x;
    xs[b] = fmaf(t2[b], c2[0], c2[1]);
}

// ---------------- Pass B: leaf routing sums S0/S1 ----------------
// Block handles 64 batch samples; full 4-tree expansion in LDS; per-thread
// register accumulators for the 4*2048 leaf sums, flushed with atomics.
__global__ __launch_bounds__(256) void k_leaf_stats(const float* __restrict__ xs,
                                                    const float* __restrict__ yb,
                                                    const float* __restrict__ tw,
                                                    const float* __restrict__ tbv,
                                                    float* __restrict__ S0,
                                                    float* __restrict__ S1) {
    __shared__ float mu[NTREES][4096];  // node-indexed 0..4095 (64 KB)
    int tid = threadIdx.x;
    __builtin_prefetch(tw + tid * 32, 0, 0);
    __builtin_prefetch(tbv + tid * 32, 0, 0);
    float2 acc[32];
#pragma unroll
    for (int j = 0; j < 32; ++j) acc[j] = make_float2(0.0f, 0.0f);

    for (int nb = 0; nb < 64; ++nb) {
        int b = blockIdx.x * 64 + nb;
        float xv = xs[b];
        float yv = yb[b];
        for (int k = 0; k < 11; ++k) {
            int width = NTREES << k;  // parents across 4 trees
            for (int p = tid; p < width; p += 256) {
                int t    = p >> k;
                int node = (1 << k) + (p & ((1 << k) - 1));
                float mup = (k == 0) ? 1.0f : mu[t][node];
                float d = sigmoidf(fmaf(xv, tw[t * NNODES + node], tbv[t * NNODES + node]));
                mu[t][2 * node]     = mup * d;
                mu[t][2 * node + 1] = mup * (1.0f - d);
            }
            __syncthreads();
        }
#pragma unroll
        for (int j = 0; j < 32; ++j) {
            int f = tid + 256 * j;        // f = t*2048 + leaf
            int t = f >> 11;
            int l = f & 2047;
            float m = mu[t][2048 + l];
            acc[j].x += m;
            acc[j].y += yv * m;
        }
        // safe: leaves only rewritten at level 10 of next sample, after 10 barriers
    }
#pragma unroll
    for (int j = 0; j < 32; ++j) {
        int f = tid + 256 * j;
        atomicAdd(&S0[f], acc[j].x);
        atomicAdd(&S1[f], acc[j].y);
    }
}

__global__ __launch_bounds__(256) void k_yhat(const float* __restrict__ S0,
                                              const float* __restrict__ S1,
                                              float* __restrict__ yh) {
    int f = blockIdx.x * 256 + threadIdx.x;
    yh[f] = S1[f] / S0[f];
}

// ---------------- Pass C: prediction via V_WMMA_F32_16X16X4_F32 ----------------
// Wave-per-16-sample tile. A(16x4) = mu leaf tile (lane L: M=L%16, K=2*(L/16)+v),
// B(4x16) = yhat in column 0 only. D accumulates over all leaves and trees.
__global__ __launch_bounds__(256) void k_pred_wmma(const float* __restrict__ xs,
                                                   const float* __restrict__ tw,
                                                   const float* __restrict__ tbv,
                                                   const float* __restrict__ yhg,
                                                   float* __restrict__ out) {
    __shared__ float yh[NTREES][NNODES];  // 32 KB
    for (int i = threadIdx.x; i < NTREES * NNODES; i += 256)
        ((float*)yh)[i] = yhg[i];
    __builtin_prefetch(tw + threadIdx.x * 32, 0, 0);
    __builtin_prefetch(tbv + threadIdx.x * 32, 0, 0);
    __syncthreads();

    int lane = threadIdx.x & 31;
    int wave = threadIdx.x >> 5;
    int tile = blockIdx.x * 8 + wave;
    int bcol = lane & 15;   // A-matrix row M (batch sample within tile)
    int kg   = lane >> 4;   // lane half selects K pair {0,1} vs {2,3}
    float xv = xs[tile * 16 + bcol];

    v8f c = {0.0f, 0.0f, 0.0f, 0.0f, 0.0f, 0.0f, 0.0f, 0.0f};
    for (int t = 0; t < NTREES; ++t) {
        const float* w  = tw + t * NNODES;
        const float* bb = tbv + t * NNODES;
        for (int j = 0; j < 512; ++j) {       // leaf chunk of 4 per WMMA
            int l0 = 4 * j + 2 * kg;          // even leaf; sibling = l0+1
            int n0 = 2048 + l0;               // leaf node id (12 bits, bit11 set)
            float mup = 1.0f;                 // path product root..level-9
#pragma unroll
            for (int k = 0; k < 10; ++k) {
                int anc = n0 >> (11 - k);
                int bit = (n0 >> (10 - k)) & 1;
                float d = sigmoidf(fmaf(xv, w[anc], bb[anc]));
                mup *= bit ? (1.0f - d) : d;
            }
            int p = n0 >> 1;                  // level-10 parent of sibling pair
            float dp = sigmoidf(fmaf(xv, w[p], bb[p]));
            v2f a;
            a.x = mup * dp;                   // leaf l0   (K = 2*kg)
            a.y = mup * (1.0f - dp);          // leaf l0+1 (K = 2*kg+1)
            v2f bv;
            bv.x = (bcol == 0) ? yh[t][l0] : 0.0f;      // B[K,0]
            bv.y = (bcol == 0) ? yh[t][l0 + 1] : 0.0f;
            c = __builtin_amdgcn_wmma_f32_16x16x4_f32(
                    false, a, false, bv, (short)0, c, false, false);
        }
    }
    // D column 0: lane 0 -> rows 0..7, lane 16 -> rows 8..15
    if (bcol == 0) {
        int mbase = tile * 16 + 8 * kg;
#pragma unroll
        for (int r = 0; r < 8; ++r) out[mbase + r] = 0.25f * c[r];
    }
}

// ---------------- launcher ----------------
extern "C" void kernel_launch(void* const* d_in, const int* in_sizes, int n_in,
                              void* d_out, int out_size, void* d_ws, size_t ws_size,
                              hipStream_t stream) {
    (void)in_sizes; (void)n_in; (void)out_size; (void)ws_size;
    const float* xb  = (const float*)d_in[0];
    const float* yb  = (const float*)d_in[1];
    const float* w1  = (const float*)d_in[2];
    const float* b1  = (const float*)d_in[3];
    const float* g1  = (const float*)d_in[4];
    const float* be1 = (const float*)d_in[5];
    const float* w2  = (const float*)d_in[6];
    const float* b2  = (const float*)d_in[7];
    const float* g2  = (const float*)d_in[8];
    const float* be2 = (const float*)d_in[9];
    const float* tw  = (const float*)d_in[10];
    const float* tbv = (const float*)d_in[11];

    float* ws     = (float*)d_ws;
    float* stats1 = ws + 0;      // 32
    float* Avec   = ws + 32;     // 16
    float* C0     = ws + 48;     // 1
    float* stats2 = ws + 49;     // 2
    float* c2     = ws + 52;     // 2
    float* t2     = ws + 64;     // 8192
    float* xsv    = ws + 8256;   // 8192
    float* S0     = ws + 16448;  // 8192
    float* S1     = ws + 24640;  // 8192
    float* yh     = ws + 32832;  // 8192

    k_zero<<<(32832 + 255) / 256, 256, 0, stream>>>(ws, 32832);
    k_bn1_stats<<<BATCH / 256, 256, 0, stream>>>(xb, w1, b1, stats1);
    k_consts1<<<1, 32, 0, stream>>>(stats1, g1, be1, w2, b2, Avec, C0);
    k_bn2_stats<<<BATCH / 256, 256, 0, stream>>>(xb, w1, b1, Avec, C0, t2, stats2);
    k_consts2<<<1, 32, 0, stream>>>(stats2, g2, be2, c2);
    k_xs<<<BATCH / 256, 256, 0, stream>>>(t2, c2, xsv);
    k_leaf_stats<<<BATCH / 64, 256, 0, stream>>>(xsv, yb, tw, tbv, S0, S1);
    k_yhat<<<(NTREES * NNODES) / 256, 256, 0, stream>>>(S0, S1, yh);
    k_pred_wmma<<<BATCH / (16 * 8), 256, 0, stream>>>(xsv, tw, tbv, yh, (float*)d_out);
}